// AttentionWeightedRetrieval_11519102288261
// MI455X (gfx1250) — compile-verified
//
#include <hip/hip_runtime.h>

#define D_DIM 64
#define C_DIM 128
#define EPSF 1e-6f
#define INV_TEMP 10.0f
#define DTF 0.1f
#define LDS_STRIDE 68          // 64 + 4 pad floats -> conflict-free ds_load_b64
#define ROWS_PER_BLOCK 128     // 8 waves x 16 rows
#define SUM_BLOCKS 512

typedef float v2f __attribute__((ext_vector_type(2)));
typedef float v8f __attribute__((ext_vector_type(8)));

__device__ __forceinline__ unsigned key_encode(float x) {
  unsigned u = __float_as_uint(x);
  return (u & 0x80000000u) ? ~u : (u | 0x80000000u);
}
__device__ __forceinline__ float key_decode(unsigned k) {
  return (k & 0x80000000u) ? __uint_as_float(k ^ 0x80000000u)
                           : __uint_as_float(~k);
}

// ---------------------------------------------------------------- kernel 1
// query field step + q_pot + scratch-scalar init (tiny: 128x64 work)
__global__ __launch_bounds__(128) void k_query(const float* __restrict__ q,
                                               const float* __restrict__ centers,
                                               const float* __restrict__ mus,
                                               float* __restrict__ out,
                                               float* __restrict__ wsf,
                                               unsigned* __restrict__ wsu) {
  __shared__ float sq[D_DIM];
  __shared__ float sw[C_DIM];
  __shared__ float spot[C_DIM];
  const int tid = threadIdx.x;              // 128 threads, one per center
  if (tid < D_DIM) sq[tid] = q[tid];
  __syncthreads();
  {
    float d2 = 0.f;
#pragma unroll 8
    for (int c = 0; c < D_DIM; ++c) {
      float df = centers[tid * D_DIM + c] - sq[c];
      d2 = __builtin_fmaf(df, df, d2);
    }
    d2 = fmaxf(d2, 1e-12f);
    float r = sqrtf(d2);
    float mu = mus[tid];
    spot[tid] = mu / (r + EPSF);            // potential term
    sw[tid] = mu / (r * r * r + EPSF);      // field weight
  }
  __syncthreads();
  if (tid == 0) {
    float p = 0.f;
    for (int j = 0; j < C_DIM; ++j) p += spot[j];
    wsf[0] = p;      // q_pot
    wsu[1] = 0u;     // softmax-max key (monotone uint encoding, 0 == -inf)
    wsf[2] = 0.f;    // softmax denominator
  }
  if (tid < D_DIM) {
    float f = 0.f;
    for (int j = 0; j < C_DIM; ++j)
      f += sw[j] * (centers[j * D_DIM + tid] - sq[tid]);
    out[tid] = sq[tid] + DTF * f;           // query_output
  }
}

// ---------------------------------------------------------------- kernel 2
// c_pot via fp32 WMMA: (16 rows x 64) . (64 x 128 centers) per wave
__global__ __launch_bounds__(256) void k_pot(const float* __restrict__ cand,
                                             const float* __restrict__ centers,
                                             const float* __restrict__ mus,
                                             const float* __restrict__ wsf,
                                             unsigned* __restrict__ wsu,
                                             float* __restrict__ logits,
                                             int n) {
  __shared__ float sC[C_DIM * LDS_STRIDE];  // padded centers (B operand)
  __shared__ float sCN[C_DIM];              // ||center||^2
  __shared__ float sMu[C_DIM];
  const int tid = threadIdx.x;

  for (int i = tid; i < C_DIM * D_DIM; i += 256) {
    int r = i >> 6, c = i & 63;
    sC[r * LDS_STRIDE + c] = centers[i];
  }
  __syncthreads();
  if (tid < C_DIM) {
    float s = 0.f;
#pragma unroll
    for (int c = 0; c < D_DIM; ++c) {
      float v = sC[tid * LDS_STRIDE + c];
      s = __builtin_fmaf(v, v, s);
    }
    sCN[tid] = s;
    sMu[tid] = mus[tid];
  }
  __syncthreads();

  const int wave = tid >> 5;
  const int lane = tid & 31;
  const int h = lane >> 4;                  // half-wave id (K / M-group select)
  const int l15 = lane & 15;
  const long base = (long)blockIdx.x * ROWS_PER_BLOCK + (long)wave * 16;
  long rowMine = base + l15;
  long rowLd = rowMine < (long)n ? rowMine : (long)(n - 1);  // clamp tail
  const float* aptr = cand + rowLd * D_DIM + 2 * h;

  v8f acc[8];
  const v8f vzero = {0.f, 0.f, 0.f, 0.f, 0.f, 0.f, 0.f, 0.f};
#pragma unroll
  for (int t = 0; t < 8; ++t) acc[t] = vzero;

  float asq = 0.f;                          // candidate-norm side product
#pragma unroll
  for (int k = 0; k < 16; ++k) {            // K = 64 in steps of 4
    float2 av = *(const float2*)(aptr + 4 * k);
    asq = __builtin_fmaf(av.x, av.x, asq);
    asq = __builtin_fmaf(av.y, av.y, asq);
    v2f af = {av.x, av.y};                  // A frag: row l15, K pair per half
#pragma unroll
    for (int t = 0; t < 8; ++t) {           // 8 column tiles of 16 centers
      float2 bv = *(const float2*)&sC[(t * 16 + l15) * LDS_STRIDE + k * 4 + 2 * h];
      v2f bf = {bv.x, bv.y};                // B frag: B[k][n] = centers[n][k]
      acc[t] = __builtin_amdgcn_wmma_f32_16x16x4_f32(
          false, af, false, bf, (short)0, acc[t], false, false);
    }
  }

  // full ||cand_row||^2: lane l and l+16 hold complementary column halves
  float rown = asq + __shfl_xor(asq, 16, 32);
  float nm[8];
#pragma unroll
  for (int v = 0; v < 8; ++v)
    nm[v] = __shfl(rown, h * 8 + v, 32);    // norm of C/D row M = 8h+v

  float part[8];
#pragma unroll
  for (int v = 0; v < 8; ++v) part[v] = 0.f;

#pragma unroll
  for (int t = 0; t < 8; ++t) {
    int j = t * 16 + l15;                   // center index for this lane
    float cn = sCN[j];
    float mu = sMu[j];
#pragma unroll
    for (int v = 0; v < 8; ++v) {
      float d2 = nm[v] + cn - 2.0f * acc[t][v];
      d2 = fmaxf(d2, 1e-12f);
      float d = sqrtf(d2);
      part[v] += mu / (d + EPSF);
    }
  }
  // reduce over the 16 N-lanes within each half-wave (h stays fixed)
#pragma unroll
  for (int v = 0; v < 8; ++v) {
#pragma unroll
    for (int off = 1; off < 16; off <<= 1)
      part[v] += __shfl_xor(part[v], off, 32);
  }

  const float q_pot = wsf[0];
  float wavemax = -3.0e38f;
#pragma unroll
  for (int v = 0; v < 8; ++v) {
    long row = base + (long)(h * 8 + v);
    if (row < (long)n) {
      float logit = -fabsf(q_pot - part[v]) * INV_TEMP;
      wavemax = fmaxf(wavemax, logit);
      if (l15 == v) logits[row] = logit;    // one writer per row
    }
  }
#pragma unroll
  for (int off = 1; off < 32; off <<= 1)
    wavemax = fmaxf(wavemax, __shfl_xor(wavemax, off, 32));
  if (lane == 0)
    atomicMax(wsu + 1, key_encode(wavemax)); // integer max: deterministic
}

// ---------------------------------------------------------------- kernel 3a
__global__ __launch_bounds__(256) void k_sum_partial(const float* __restrict__ logits,
                                                     const unsigned* __restrict__ wsu,
                                                     float* __restrict__ partials,
                                                     int n) {
  __shared__ float sred[256];
  const float m = key_decode(wsu[1]);
  float acc = 0.f;
  for (long i = (long)blockIdx.x * 256 + threadIdx.x; i < (long)n;
       i += (long)gridDim.x * 256)
    acc += expf(logits[i] - m);
  sred[threadIdx.x] = acc;
  __syncthreads();
  for (int s = 128; s > 0; s >>= 1) {
    if (threadIdx.x < s) sred[threadIdx.x] += sred[threadIdx.x + s];
    __syncthreads();
  }
  if (threadIdx.x == 0) partials[blockIdx.x] = sred[0];
}

// ---------------------------------------------------------------- kernel 3b
__global__ __launch_bounds__(256) void k_sum_final(const float* __restrict__ partials,
                                                   float* __restrict__ wsf) {
  __shared__ float sred[256];
  sred[threadIdx.x] = partials[threadIdx.x] + partials[threadIdx.x + 256];
  __syncthreads();
  for (int s = 128; s > 0; s >>= 1) {
    if (threadIdx.x < s) sred[threadIdx.x] += sred[threadIdx.x + s];
    __syncthreads();
  }
  if (threadIdx.x == 0) wsf[2] = sred[0];
}

// ---------------------------------------------------------------- kernel 4
__global__ __launch_bounds__(256) void k_norm(float* __restrict__ att,
                                              const unsigned* __restrict__ wsu,
                                              const float* __restrict__ wsf,
                                              int n) {
  long i = (long)blockIdx.x * 256 + threadIdx.x;
  if (i < (long)n) {
    float m = key_decode(wsu[1]);
    float inv = 1.0f / wsf[2];
    att[i] = expf(att[i] - m) * inv;        // in-place logits -> attention
  }
}

extern "C" void kernel_launch(void* const* d_in, const int* in_sizes, int n_in,
                              void* d_out, int out_size, void* d_ws, size_t ws_size,
                              hipStream_t stream) {
  const float* q = (const float*)d_in[0];
  const float* cand = (const float*)d_in[1];
  const float* centers = (const float*)d_in[2];
  const float* mus = (const float*)d_in[3];
  float* out = (float*)d_out;
  float* wsf = (float*)d_ws;
  unsigned* wsu = (unsigned*)d_ws;
  float* partials = wsf + 16;
  const int n = in_sizes[1] / D_DIM;        // 500000
  float* logits = out + D_DIM;              // reuse output tail as scratch

  k_query<<<1, 128, 0, stream>>>(q, centers, mus, out, wsf, wsu);
  const int nblk = (n + ROWS_PER_BLOCK - 1) / ROWS_PER_BLOCK;
  k_pot<<<nblk, 256, 0, stream>>>(cand, centers, mus, wsf, wsu, logits, n);
  k_sum_partial<<<SUM_BLOCKS, 256, 0, stream>>>(logits, wsu, partials, n);
  k_sum_final<<<1, 256, 0, stream>>>(partials, wsf);
  k_norm<<<(n + 255) / 256, 256, 0, stream>>>(logits, wsu, wsf, n);
}